// DynamicConv_15126874816537
// MI455X (gfx1250) — compile-verified
//
#include <hip/hip_runtime.h>
#include <hip/hip_bf16.h>
#include <cstdint>

#define B_ 8
#define C_ 512
#define T_ 2048
#define HEADS_ 8
#define KER_ 31
#define NROWS (B_ * T_)
#define EPS_ 1e-5f

typedef __attribute__((ext_vector_type(2))) float v2f;
typedef __attribute__((ext_vector_type(8))) float v8f;

__device__ __forceinline__ v8f wmma_f32(v2f a, v2f b, v8f c) {
  // D = A(16x4 f32) * B(4x16 f32) + C(16x16 f32)
  return __builtin_amdgcn_wmma_f32_16x16x4_f32(false, a, false, b, (short)0, c,
                                               false, false);
}

// ---------------------------------------------------------------------------
// Kernel 0: LayerNorm statistics per (b,t) row. stats[row] = {mu, rstd}
// ---------------------------------------------------------------------------
__global__ __launch_bounds__(256) void ln_stats_kernel(
    const float* __restrict__ x, float* __restrict__ stats) {
  const int tid = threadIdx.x;
  const int row0 = blockIdx.x * 64;
  const int b = row0 / T_;
  const int t0 = row0 % T_;
  const int r = tid & 63;   // row within tile (consecutive t -> coalesced)
  const int g = tid >> 6;   // 4 channel groups of 128
  float s = 0.f, s2 = 0.f;
  const float* xb = x + (size_t)b * C_ * T_ + t0 + r;
#pragma unroll 4
  for (int i = 0; i < 128; ++i) {
    int c = g * 128 + i;
    float v = xb[(size_t)c * T_];
    s += v;
    s2 += v * v;
  }
  __shared__ float sb[4][64], sq[4][64];
  sb[g][r] = s;
  sq[g][r] = s2;
  __syncthreads();
  if (tid < 64) {
    float S = sb[0][tid] + sb[1][tid] + sb[2][tid] + sb[3][tid];
    float S2 = sq[0][tid] + sq[1][tid] + sq[2][tid] + sq[3][tid];
    float mu = S * (1.f / C_);
    float var = S2 * (1.f / C_) - mu * mu;
    stats[(row0 + tid) * 2 + 0] = mu;
    stats[(row0 + tid) * 2 + 1] = rsqrtf(var + EPS_);
  }
}

// ---------------------------------------------------------------------------
// Kernel A: fused LN-apply + GEMM1 (h @ Wk + bk) + GLU, k stored as (B,C,T)
// WG tile: 64 rows x 64 k-columns (needs proj cols [n0,n0+64) and +512)
// ---------------------------------------------------------------------------
#define AS_S 66
#define BS_S 66
#define KT_S 65
__global__ __launch_bounds__(256) void glu_gemm_kernel(
    const float* __restrict__ x, const float* __restrict__ gamma,
    const float* __restrict__ beta, const float* __restrict__ Wk,
    const float* __restrict__ bk, const float* __restrict__ stats,
    float* __restrict__ k_ws) {
  __shared__ float smem[64 * AS_S + 128 * BS_S];
  float* As = smem;                 // [64 rows][66] layernormed x chunk
  float* Bs = smem + 64 * AS_S;     // [128 n][66] Wk chunk, n-major (transposed)

  const int tid = threadIdx.x;
  const int row0 = blockIdx.x * 64;
  const int n0 = blockIdx.y * 64;
  const int b = row0 / T_;
  const int t0 = row0 % T_;
  const int r_ld = tid & 63;
  const int kg = tid >> 6;
  const float mu = stats[(row0 + r_ld) * 2 + 0];
  const float rs = stats[(row0 + r_ld) * 2 + 1];

  const int lane = tid & 31;
  const int wave = tid >> 5;
  const int laneM = lane & 15;
  const int hi = lane >> 4;
  const int rblk = wave & 3;        // 4 row blocks of 16
  const int j0 = (wave >> 2) * 32;  // 2 column halves of 32

  v8f accA0 = {}, accA1 = {}, accB0 = {}, accB1 = {};
  const float* xb = x + (size_t)b * C_ * T_ + t0 + r_ld;

  for (int kbase = 0; kbase < C_; kbase += 64) {
    // stage A: layernorm applied on the fly, coalesced along t
#pragma unroll
    for (int i = 0; i < 16; ++i) {
      int kk = kg * 16 + i;
      int c = kbase + kk;
      float v = xb[(size_t)c * T_];
      As[r_ld * AS_S + kk] = (v - mu) * rs * gamma[c] + beta[c];
    }
    // stage B: Wk columns for a-part and b-part, stored n-major
    {
      int jj = tid & 63;
      int kq = tid >> 6;
#pragma unroll
      for (int part = 0; part < 2; ++part) {
        int n = n0 + jj + part * C_;
#pragma unroll
        for (int i = 0; i < 16; ++i) {
          int kk = kq * 16 + i;
          Bs[(part * 64 + jj) * BS_S + kk] =
              Wk[(size_t)(kbase + kk) * (2 * C_) + n];
        }
      }
    }
    __syncthreads();
#pragma unroll
    for (int kb = 0; kb < 64; kb += 4) {
      v2f af = *reinterpret_cast<const v2f*>(
          &As[(rblk * 16 + laneM) * AS_S + kb + 2 * hi]);
      v2f b0 = *reinterpret_cast<const v2f*>(
          &Bs[(j0 + laneM) * BS_S + kb + 2 * hi]);
      v2f b1 = *reinterpret_cast<const v2f*>(
          &Bs[(j0 + 16 + laneM) * BS_S + kb + 2 * hi]);
      v2f b2 = *reinterpret_cast<const v2f*>(
          &Bs[(64 + j0 + laneM) * BS_S + kb + 2 * hi]);
      v2f b3 = *reinterpret_cast<const v2f*>(
          &Bs[(64 + j0 + 16 + laneM) * BS_S + kb + 2 * hi]);
      accA0 = wmma_f32(af, b0, accA0);
      accA1 = wmma_f32(af, b1, accA1);
      accB0 = wmma_f32(af, b2, accB0);
      accB1 = wmma_f32(af, b3, accB1);
    }
    __syncthreads();
  }

  // GLU: k = (a + bk_a) * sigmoid(b + bk_b); transpose via LDS, store (B,C,T)
  float* kst = smem;  // reuse: [64 c][65 t]
  const float bka0 = bk[n0 + j0 + laneM];
  const float bka1 = bk[n0 + j0 + 16 + laneM];
  const float bkb0 = bk[C_ + n0 + j0 + laneM];
  const float bkb1 = bk[C_ + n0 + j0 + 16 + laneM];
#pragma unroll
  for (int v = 0; v < 8; ++v) {
    int tl = rblk * 16 + v + 8 * hi;  // C/D layout: M = v + 8*hi
    float a0 = accA0[v] + bka0;
    float g0 = accB0[v] + bkb0;
    float a1 = accA1[v] + bka1;
    float g1 = accB1[v] + bkb1;
    kst[(j0 + laneM) * KT_S + tl] = a0 / (1.f + __expf(-g0));
    kst[(j0 + 16 + laneM) * KT_S + tl] = a1 / (1.f + __expf(-g1));
  }
  __syncthreads();
  float* kout = k_ws + (size_t)b * C_ * T_;
  for (int i = tid; i < 64 * 64; i += 256) {
    int cc = i >> 6;
    int tl = i & 63;
    kout[(size_t)(n0 + cc) * T_ + t0 + tl] = kst[cc * KT_S + tl];
  }
}

// ---------------------------------------------------------------------------
// Kernel B: GEMM2 (k @ Wc + bc) + per-head softmax -> w (row-stride 256)
// WG tile: 32 rows x 256 cols (N padded 248->256)
// ---------------------------------------------------------------------------
#define A2_S 34
#define B2_S 34
#define P_S 257
__global__ __launch_bounds__(256) void conv_weight_kernel(
    const float* __restrict__ k_ws, const float* __restrict__ Wc,
    const float* __restrict__ bc, float* __restrict__ w_ws) {
  __shared__ float As[32 * A2_S];
  __shared__ float Bs[256 * B2_S];
  __shared__ float proj[32 * P_S];
  const int tid = threadIdx.x;
  const int row0 = blockIdx.x * 32;
  const int b = row0 / T_;
  const int t0 = row0 % T_;
  const int lane = tid & 31;
  const int wave = tid >> 5;
  const int laneM = lane & 15;
  const int hi = lane >> 4;
  const int rb = wave & 1;          // 2 row blocks
  const int jb = (wave >> 1) * 64;  // 4 column quads of 64

  v8f acc[4] = {{}, {}, {}, {}};
  const float* kin = k_ws + (size_t)b * C_ * T_ + t0;

  for (int kbase = 0; kbase < C_; kbase += 32) {
    {  // A from k (B,C,T): coalesced along t
      int r = tid & 31;
      int cq = tid >> 5;
#pragma unroll
      for (int i = 0; i < 4; ++i) {
        int kk = cq + 8 * i;
        As[r * A2_S + kk] = kin[(size_t)(kbase + kk) * T_ + r];
      }
    }
    {  // B from Wc (C x 248), zero-pad to 256, stored n-major
      int nn = tid & 63;
      int kq = tid >> 6;
#pragma unroll
      for (int q = 0; q < 4; ++q) {
        int n = nn + 64 * q;
#pragma unroll
        for (int i = 0; i < 8; ++i) {
          int kk = kq * 8 + i;
          float v = (n < KER_ * HEADS_)
                        ? Wc[(size_t)(kbase + kk) * (KER_ * HEADS_) + n]
                        : 0.f;
          Bs[n * B2_S + kk] = v;
        }
      }
    }
    __syncthreads();
#pragma unroll
    for (int kb = 0; kb < 32; kb += 4) {
      v2f af = *reinterpret_cast<const v2f*>(
          &As[(rb * 16 + laneM) * A2_S + kb + 2 * hi]);
#pragma unroll
      for (int jt = 0; jt < 4; ++jt) {
        v2f bf = *reinterpret_cast<const v2f*>(
            &Bs[(jb + jt * 16 + laneM) * B2_S + kb + 2 * hi]);
        acc[jt] = wmma_f32(af, bf, acc[jt]);
      }
    }
    __syncthreads();
  }
  // spill accumulators to LDS for the softmax pass
#pragma unroll
  for (int jt = 0; jt < 4; ++jt)
#pragma unroll
    for (int v = 0; v < 8; ++v)
      proj[(rb * 16 + v + 8 * hi) * P_S + jb + jt * 16 + laneM] = acc[jt][v];
  __syncthreads();

  // softmax: one thread per (row, head); 32 rows x 8 heads = 256 threads
  {
    int rr = tid >> 3;
    int h = tid & 7;
    float* prow = &proj[rr * P_S + h * KER_];
    const float* bch = &bc[h * KER_];
    float m = -1e30f;
#pragma unroll
    for (int i = 0; i < KER_; ++i) {
      float v = prow[i] + bch[i];
      prow[i] = v;
      m = fmaxf(m, v);
    }
    float s = 0.f;
#pragma unroll
    for (int i = 0; i < KER_; ++i) {
      float e = __expf(prow[i] - m);
      prow[i] = e;
      s += e;
    }
    float inv = 1.f / s;
    float* wout = w_ws + (size_t)(row0 + rr) * 256 + h * 32;
#pragma unroll
    for (int i = 0; i < KER_; ++i) wout[i] = prow[i] * inv;
    wout[KER_] = 0.f;  // pad slot
  }
}

// ---------------------------------------------------------------------------
// Kernel C: dynamic conv + conv_bias + residual, output (B,C,T)
// block = (b, 32 t-positions); threads: 32 t x 8 heads; 64 channels per head
// ---------------------------------------------------------------------------
__global__ __launch_bounds__(256) void dynconv_kernel(
    const float* __restrict__ x, const float* __restrict__ k_ws,
    const float* __restrict__ w_ws, const float* __restrict__ conv_bias,
    float* __restrict__ out) {
  __shared__ float wl[32 * 257];  // odd stride -> conflict-free per-t reads
  const int tid = threadIdx.x;
  const int t0 = blockIdx.x * 32;
  const int b = blockIdx.y;
  const float* wsrc = w_ws + ((size_t)b * T_ + t0) * 256;
  for (int i = tid; i < 32 * 256; i += 256)
    wl[(i >> 8) * 257 + (i & 255)] = wsrc[i];
  __syncthreads();

  const int tx = tid & 31;  // t within tile (coalesced dimension)
  const int cg = tid >> 5;  // head (8 heads x 64 channels)
  const int t = t0 + tx;
  const float* wrow = &wl[tx * 257 + cg * 32];
  const bool interior = (t0 >= KER_ / 2) && (t0 + 32 + KER_ / 2 <= T_);

#pragma unroll 1
  for (int ci = 0; ci < 64; ++ci) {
    int c = cg * 64 + ci;
    const size_t basec = ((size_t)b * C_ + c) * T_;
    float acc = conv_bias[c] + x[basec + t];
    if (interior) {
#pragma unroll
      for (int kk = 0; kk < KER_; ++kk)
        acc = fmaf(k_ws[basec + t + kk - KER_ / 2], wrow[kk], acc);
    } else {
#pragma unroll
      for (int kk = 0; kk < KER_; ++kk) {
        int tt = t + kk - KER_ / 2;
        if ((unsigned)tt < (unsigned)T_)
          acc = fmaf(k_ws[basec + tt], wrow[kk], acc);
      }
    }
    out[basec + t] = acc;
  }
}

// ---------------------------------------------------------------------------
extern "C" void kernel_launch(void* const* d_in, const int* in_sizes, int n_in,
                              void* d_out, int out_size, void* d_ws,
                              size_t ws_size, hipStream_t stream) {
  (void)in_sizes;
  (void)n_in;
  (void)out_size;
  (void)ws_size;
  const float* x = (const float*)d_in[0];
  const float* ln_gamma = (const float*)d_in[1];
  const float* ln_beta = (const float*)d_in[2];
  const float* Wk = (const float*)d_in[3];
  const float* bk = (const float*)d_in[4];
  const float* Wc = (const float*)d_in[5];
  const float* bc = (const float*)d_in[6];
  const float* conv_bias = (const float*)d_in[7];
  float* out = (float*)d_out;

  float* ws = (float*)d_ws;
  float* stats = ws;                        // 32768 floats
  float* k_ws = ws + 32768;                 // 8,388,608 floats (B,C,T)
  float* w_ws = k_ws + (size_t)B_ * C_ * T_;  // 4,194,304 floats

  ln_stats_kernel<<<NROWS / 64, 256, 0, stream>>>(x, stats);
  glu_gemm_kernel<<<dim3(NROWS / 64, C_ / 64), 256, 0, stream>>>(
      x, ln_gamma, ln_beta, Wk, bk, stats, k_ws);
  conv_weight_kernel<<<NROWS / 32, 256, 0, stream>>>(k_ws, Wc, bc, w_ws);
  dynconv_kernel<<<dim3(T_ / 32, B_), 256, 0, stream>>>(x, k_ws, w_ws,
                                                        conv_bias, out);
}